// XformerDecoderLayer_22445499089634
// MI455X (gfx1250) — compile-verified
//
#include <hip/hip_runtime.h>
#include <math.h>

typedef float v2f __attribute__((ext_vector_type(2)));
typedef float v8f __attribute__((ext_vector_type(8)));

#define D_MODEL 256
#define NHEAD   8
#define DHEAD   32
#define DFF     2048
#define BATCH   32
#define NQ      300
#define LTGT    (BATCH * NQ)   // 9600
#define LMEM    24576

// -------- fp32 WMMA wrapper: D = A(16x4) * B(4x16) + C, exact fp32 FMA --------
__device__ __forceinline__ v8f wmma4(v2f a, v2f b, v8f c) {
  return __builtin_amdgcn_wmma_f32_16x16x4_f32(false, a, false, b, (short)0, c,
                                               false, false);
}

// =====================================================================
// GEMM: Y[M,N] = A1[M,K] @ W1[N,K]^T (+ A2 @ W2^T) + b1 (+ b2) (+ res), relu?
// block = 256 threads = 8 waves arranged 2x4; wave tile 64x32
// (4 A-frags x 2 B-frags -> 8 WMMAs per 6 float2 loads); block tile 128x128.
// FUSE2 is compile-time so the inner loop is branch-free.
// =====================================================================
template <bool FUSE2>
__global__ void __launch_bounds__(256)
gemm_kernel(const float* __restrict__ A1, const float* __restrict__ W1,
            const float* __restrict__ A2, const float* __restrict__ W2,
            const float* __restrict__ b1, const float* __restrict__ b2,
            const float* __restrict__ res, float* __restrict__ Y,
            int M, int N, int K, int do_relu)
{
  const int lane = threadIdx.x & 31;
  const int wave = threadIdx.x >> 5;
  const int lo   = lane & 15;
  const int hi   = lane >> 4;          // 0 / 1 : lane half
  const int r0   = blockIdx.y * 128 + (wave >> 2) * 64;
  const int c0   = blockIdx.x * 128 + (wave & 3) * 32;

  v8f acc[4][2] = {};

  size_t rowA[4], rowW[2];
#pragma unroll
  for (int r = 0; r < 4; ++r)
    rowA[r] = (size_t)(r0 + 16 * r + lo) * (size_t)K;
  rowW[0] = (size_t)(c0 + lo)      * (size_t)K;
  rowW[1] = (size_t)(c0 + 16 + lo) * (size_t)K;
  const int koff = 2 * hi;

  for (int k = 0; k < K; k += 4) {
    v2f w0 = *(const v2f*)(W1 + rowW[0] + k + koff);
    v2f w1 = *(const v2f*)(W1 + rowW[1] + k + koff);
    v2f a[4];
#pragma unroll
    for (int r = 0; r < 4; ++r)
      a[r] = *(const v2f*)(A1 + rowA[r] + k + koff);
#pragma unroll
    for (int r = 0; r < 4; ++r) {
      acc[r][0] = wmma4(a[r], w0, acc[r][0]);
      acc[r][1] = wmma4(a[r], w1, acc[r][1]);
    }
    if constexpr (FUSE2) {   // fused second operand pair: + A2 @ W2^T
      v2f w0b = *(const v2f*)(W2 + rowW[0] + k + koff);
      v2f w1b = *(const v2f*)(W2 + rowW[1] + k + koff);
      v2f ab[4];
#pragma unroll
      for (int r = 0; r < 4; ++r)
        ab[r] = *(const v2f*)(A2 + rowA[r] + k + koff);
#pragma unroll
      for (int r = 0; r < 4; ++r) {
        acc[r][0] = wmma4(ab[r], w0b, acc[r][0]);
        acc[r][1] = wmma4(ab[r], w1b, acc[r][1]);
      }
    }
  }

  float bc0 = b1[c0 + lo];
  float bc1 = b1[c0 + 16 + lo];
  if (b2) { bc0 += b2[c0 + lo]; bc1 += b2[c0 + 16 + lo]; }

#pragma unroll
  for (int r = 0; r < 4; ++r) {
#pragma unroll
    for (int i = 0; i < 8; ++i) {
      int row = r0 + 16 * r + i + 8 * hi;
      size_t base = (size_t)row * (size_t)N;
      float y0 = acc[r][0][i] + bc0;
      float y1 = acc[r][1][i] + bc1;
      if (res) {
        y0 += res[base + c0 + lo];
        y1 += res[base + c0 + 16 + lo];
      }
      if (do_relu) { y0 = fmaxf(y0, 0.f); y1 = fmaxf(y1, 0.f); }
      Y[base + c0 + lo]      = y0;
      Y[base + c0 + 16 + lo] = y1;
    }
  }
}

// =====================================================================
// LayerNorm, one wave per 256-wide row; block = 256 threads = 8 rows.
// =====================================================================
__global__ void __launch_bounds__(256)
ln_kernel(const float* __restrict__ x, const float* __restrict__ g,
          const float* __restrict__ b, float* __restrict__ y)
{
  const int wave = threadIdx.x >> 5;
  const int lane = threadIdx.x & 31;
  const size_t row = (size_t)blockIdx.x * 8 + wave;
  const float* xr = x + row * D_MODEL;
  float vals[8];
  float s = 0.f, s2 = 0.f;
#pragma unroll
  for (int j = 0; j < 8; ++j) {
    float v = xr[lane + j * 32];
    vals[j] = v; s += v; s2 += v * v;
  }
  for (int off = 16; off; off >>= 1) {
    s  += __shfl_xor(s,  off);
    s2 += __shfl_xor(s2, off);
  }
  float mu  = s * (1.0f / D_MODEL);
  float var = s2 * (1.0f / D_MODEL) - mu * mu;
  float rs  = rsqrtf(var + 1e-5f);
  float* yr = y + row * D_MODEL;
#pragma unroll
  for (int j = 0; j < 8; ++j) {
    int c = lane + j * 32;
    yr[c] = (vals[j] - mu) * rs * g[c] + b[c];
  }
}

// =====================================================================
// Flash attention (online softmax), one wave per 16-query tile.
// S = (Qa.Ka^T [+ Qb.Kb^T]) * scale (scale folded into Q fragments),
// softmax over ragged kv length, O = P @ V.  P tile goes C-layout -> LDS
// -> A-fragment; row sums of P are computed on the matrix pipe as P @ 1
// (C-layout result matches the per-row l accumulator exactly).
// HAS_QB compile-time -> branch-free kv loop.
// grid = (19, NHEAD, BATCH), block = 32 (one wave).
// =====================================================================
template <bool HAS_QB>
__global__ void __launch_bounds__(32)
attn_kernel(const float* __restrict__ Qa, const float* __restrict__ Qb,
            const float* __restrict__ Ka, const float* __restrict__ Kb,
            const float* __restrict__ V,  float* __restrict__ O,
            const int* __restrict__ lens, float scale)
{
  __shared__ float Pt[16 * 16];
  const int lane = threadIdx.x & 31;
  const int lo   = lane & 15;
  const int hi   = lane >> 4;
  const int b    = blockIdx.z;
  const int h    = blockIdx.y;
  const int qt   = blockIdx.x;
  const int hd   = h * DHEAD;

  int kvbase, len;
  if (lens) {
    int off = 0;
    for (int i = 0; i < b; ++i) off += lens[i];
    kvbase = off; len = lens[b];
  } else {
    kvbase = b * NQ; len = NQ;
  }

  // Preload Q A-fragments, pre-scaled (reused across all kv tiles)
  const int qi_l = qt * 16 + lo;
  const size_t qrow =
      (size_t)(b * NQ + (qi_l < NQ ? qi_l : NQ - 1)) * D_MODEL + hd;
  v2f aq[8], aqs[8];
#pragma unroll
  for (int j = 0; j < 8; ++j) {
    aq[j] = *(const v2f*)(Qa + qrow + j * 4 + 2 * hi);
    aq[j] *= scale;
  }
  if constexpr (HAS_QB) {
#pragma unroll
    for (int j = 0; j < 8; ++j) {
      aqs[j] = *(const v2f*)(Qb + qrow + j * 4 + 2 * hi);
      aqs[j] *= scale;
    }
  }

  float m[8];
#pragma unroll
  for (int i = 0; i < 8; ++i) m[i] = -INFINITY;
  v8f lacc = {};                 // row sums, C-layout (P @ ones)
  v8f oa0 = {}, oa1 = {};
  const v2f onesv = {1.0f, 1.0f};

  const int nt = (len + 15) >> 4;
  for (int ct = 0; ct < nt; ++ct) {
    const int c0  = ct * 16;
    const int cc  = c0 + lo;
    const int kvc = kvbase + (cc < len ? cc : len - 1);
    const size_t krow = (size_t)kvc * D_MODEL + hd;

    v8f s = {};
#pragma unroll
    for (int j = 0; j < 8; ++j) {
      v2f bk = *(const v2f*)(Ka + krow + j * 4 + 2 * hi);
      s = wmma4(aq[j], bk, s);
    }
    if constexpr (HAS_QB) {
#pragma unroll
      for (int j = 0; j < 8; ++j) {
        v2f bk = *(const v2f*)(Kb + krow + j * 4 + 2 * hi);
        s = wmma4(aqs[j], bk, s);
      }
    }

    const bool valid = cc < len;
#pragma unroll
    for (int i = 0; i < 8; ++i) {
      float sv = valid ? s[i] : -INFINITY;
      float tm = sv;                       // row max across 16-lane half
      tm = fmaxf(tm, __shfl_xor(tm, 1, 16));
      tm = fmaxf(tm, __shfl_xor(tm, 2, 16));
      tm = fmaxf(tm, __shfl_xor(tm, 4, 16));
      tm = fmaxf(tm, __shfl_xor(tm, 8, 16));
      float mn   = fmaxf(m[i], tm);
      float corr = __expf(m[i] - mn);      // 0 on first tile (m = -inf)
      float p    = __expf(sv - mn);        // 0 for masked cols
      m[i] = mn;
      lacc[i] *= corr;
      oa0[i]  *= corr;
      oa1[i]  *= corr;
      Pt[(i + 8 * hi) * 16 + lo] = p;      // C-layout -> LDS
    }
    __syncthreads();                        // single-wave block: cheap fence

    // O += P(16x16) @ V(16x32) and l += P @ 1, P in A-fragment layout
#pragma unroll
    for (int j = 0; j < 4; ++j) {
      const int kk = j * 4 + 2 * hi;
      v2f ap;
      ap.x = Pt[lo * 16 + kk];
      ap.y = Pt[lo * 16 + kk + 1];
      int cA = c0 + kk;
      int cB = c0 + kk + 1;
      const float* v0 =
          V + (size_t)(kvbase + (cA < len ? cA : len - 1)) * D_MODEL + hd;
      const float* v1 =
          V + (size_t)(kvbase + (cB < len ? cB : len - 1)) * D_MODEL + hd;
      v2f bv0, bv1;
      bv0.x = v0[lo];      bv0.y = v1[lo];
      bv1.x = v0[16 + lo]; bv1.y = v1[16 + lo];
      lacc = wmma4(ap, onesv, lacc);       // row sums on the matrix pipe
      oa0  = wmma4(ap, bv0, oa0);
      oa1  = wmma4(ap, bv1, oa1);
    }
    __syncthreads();
  }

#pragma unroll
  for (int i = 0; i < 8; ++i) {
    int qi = qt * 16 + i + 8 * hi;
    if (qi < NQ) {
      float inv = 1.0f / lacc[i];
      size_t row = (size_t)(b * NQ + qi) * D_MODEL + hd;
      O[row + lo]      = oa0[i] * inv;
      O[row + 16 + lo] = oa1[i] * inv;
    }
  }
}

// =====================================================================
// Orchestration
// =====================================================================
extern "C" void kernel_launch(void* const* d_in, const int* in_sizes, int n_in,
                              void* d_out, int out_size, void* d_ws,
                              size_t ws_size, hipStream_t stream)
{
  (void)in_sizes; (void)n_in; (void)out_size; (void)ws_size;
  auto F = [&](int i) { return (const float*)d_in[i]; };

  const float* memory = F(0);
  const float* pos    = F(1);
  const float* tgt    = F(2);
  const float* qpos   = F(3);
  const float* qsine  = F(4);
  // params, dict insertion order, w then b (g then b for LN)
  const float *sa_qc_w = F(5),  *sa_qc_b = F(6),  *sa_qp_w = F(7),  *sa_qp_b = F(8);
  const float *sa_kc_w = F(9),  *sa_kc_b = F(10), *sa_kp_w = F(11), *sa_kp_b = F(12);
  const float *sa_v_w  = F(13), *sa_v_b  = F(14), *sa_o_w  = F(15), *sa_o_b  = F(16);
  const float *ca_qc_w = F(17), *ca_qc_b = F(18), *ca_qp_w = F(19), *ca_qp_b = F(20);
  const float *ca_kc_w = F(21), *ca_kc_b = F(22), *ca_kp_w = F(23), *ca_kp_b = F(24);
  const float *ca_v_w  = F(25), *ca_v_b  = F(26), *ca_qs_w = F(27), *ca_qs_b = F(28);
  const float *ca_o_w  = F(29), *ca_o_b  = F(30);
  const float *lin1_w  = F(31), *lin1_b  = F(32), *lin2_w  = F(33), *lin2_b  = F(34);
  const float *ln1_g   = F(35), *ln1_b   = F(36), *ln2_g   = F(37), *ln2_b   = F(38);
  const float *ln3_g   = F(39), *ln3_b   = F(40);
  const int*   mem_lens = (const int*)d_in[41];

  float* ws = (float*)d_ws;
  const size_t BIGF = (size_t)LTGT * DFF;          // 19,660,800 floats
  float* kc = ws;                                   // [LMEM,256]
  float* kp = ws + (size_t)LMEM * D_MODEL;
  float* vm = ws + 2 * (size_t)LMEM * D_MODEL;
  float* hf = ws;                                   // FFN hidden, reuses kc/kp/vm
  float* bufA = ws + BIGF;
  float* bufB = bufA + (size_t)LTGT * D_MODEL;
  float* bufC = bufB + (size_t)LTGT * D_MODEL;
  float* bufD = bufC + (size_t)LTGT * D_MODEL;
  float* out  = (float*)d_out;

  dim3 blk(256);
  dim3 gTgt(D_MODEL / 128, LTGT / 128);  // (2,75)
  dim3 gMem(D_MODEL / 128, LMEM / 128);  // (2,192)
  dim3 gFF1(DFF / 128, LTGT / 128);      // (16,75)
  dim3 gAttn((NQ + 15) / 16, NHEAD, BATCH);
  const float sa_scale = 0.17677669529663687f;     // 1/sqrt(32)
  const float ca_scale = 0.125f;                   // 1/sqrt(64)

  // ---- self attention ----
  ln_kernel<<<LTGT / 8, blk, 0, stream>>>(tgt, ln1_g, ln1_b, bufA);       // t1
  gemm_kernel<true><<<gTgt, blk, 0, stream>>>(bufA, sa_qc_w, qpos, sa_qp_w,
      sa_qc_b, sa_qp_b, nullptr, bufB, LTGT, D_MODEL, D_MODEL, 0);        // q
  gemm_kernel<true><<<gTgt, blk, 0, stream>>>(bufA, sa_kc_w, qpos, sa_kp_w,
      sa_kc_b, sa_kp_b, nullptr, bufC, LTGT, D_MODEL, D_MODEL, 0);        // k
  gemm_kernel<false><<<gTgt, blk, 0, stream>>>(bufA, sa_v_w, nullptr, nullptr,
      sa_v_b, nullptr, nullptr, bufD, LTGT, D_MODEL, D_MODEL, 0);         // v
  attn_kernel<false><<<gAttn, dim3(32), 0, stream>>>(bufB, nullptr, bufC,
      nullptr, bufD, bufA, nullptr, sa_scale);                            // o
  gemm_kernel<false><<<gTgt, blk, 0, stream>>>(bufA, sa_o_w, nullptr, nullptr,
      sa_o_b, nullptr, tgt, bufB, LTGT, D_MODEL, D_MODEL, 0);             // tgt1

  // ---- cross attention ----
  ln_kernel<<<LTGT / 8, blk, 0, stream>>>(bufB, ln2_g, ln2_b, bufA);      // t2
  gemm_kernel<true><<<gTgt, blk, 0, stream>>>(bufA, ca_qc_w, qpos, ca_qp_w,
      ca_qc_b, ca_qp_b, nullptr, bufC, LTGT, D_MODEL, D_MODEL, 0);        // q2
  gemm_kernel<false><<<gTgt, blk, 0, stream>>>(qsine, ca_qs_w, nullptr, nullptr,
      ca_qs_b, nullptr, nullptr, bufD, LTGT, D_MODEL, D_MODEL, 0);        // qs2
  gemm_kernel<false><<<gMem, blk, 0, stream>>>(memory, ca_kc_w, nullptr, nullptr,
      ca_kc_b, nullptr, nullptr, kc, LMEM, D_MODEL, D_MODEL, 0);          // kc
  gemm_kernel<false><<<gMem, blk, 0, stream>>>(pos, ca_kp_w, nullptr, nullptr,
      ca_kp_b, nullptr, nullptr, kp, LMEM, D_MODEL, D_MODEL, 0);          // kp
  gemm_kernel<false><<<gMem, blk, 0, stream>>>(memory, ca_v_w, nullptr, nullptr,
      ca_v_b, nullptr, nullptr, vm, LMEM, D_MODEL, D_MODEL, 0);           // vm
  attn_kernel<true><<<gAttn, dim3(32), 0, stream>>>(bufC, bufD, kc, kp, vm,
      bufA, mem_lens, ca_scale);                                          // o2
  gemm_kernel<false><<<gTgt, blk, 0, stream>>>(bufA, ca_o_w, nullptr, nullptr,
      ca_o_b, nullptr, bufB, bufC, LTGT, D_MODEL, D_MODEL, 0);            // tgt2

  // ---- FFN ----
  ln_kernel<<<LTGT / 8, blk, 0, stream>>>(bufC, ln3_g, ln3_b, bufA);      // t3
  gemm_kernel<false><<<gFF1, blk, 0, stream>>>(bufA, lin1_w, nullptr, nullptr,
      lin1_b, nullptr, nullptr, hf, LTGT, DFF, D_MODEL, 1);               // h=relu
  gemm_kernel<false><<<gTgt, blk, 0, stream>>>(hf, lin2_w, nullptr, nullptr,
      lin2_b, nullptr, bufC, out, LTGT, D_MODEL, DFF, 0);                 // out
}